// OpponentModelOracle_45449343926475
// MI455X (gfx1250) — compile-verified
//
#include <hip/hip_runtime.h>

#define BATCH 512
#define HDIM  128
#define WDIM  128
#define HW    (HDIM * WDIM)       // 16384 cells per batch
#define NWORDS (HW / 32)          // 512 bitmap words
#define TPB   256                 // 8 wave32 waves per workgroup
#define STEPS (HW / TPB)          // 64 streaming steps

__global__ __launch_bounds__(TPB)
void oracle_target_map_kernel(const float* __restrict__ x, float* __restrict__ out) {
    // LDS: double-buffered async staging (8 KB) + food bitmap (2 KB) + scalars
    __shared__ float4 stage[2][TPB];
    __shared__ unsigned int bitmap[NWORDS];
    __shared__ unsigned int s_opp;   // min row-major index with ch3==1, else ~0
    __shared__ unsigned int s_cnt;   // food count
    __shared__ unsigned int s_key;   // min (d2<<14 | idx) over food cells

    const int b    = blockIdx.x;
    const int tid  = threadIdx.x;
    const int lane = tid & 31;
    const int wave = tid >> 5;

    if (tid == 0) { s_opp = 0xFFFFFFFFu; s_cnt = 0u; s_key = 0xFFFFFFFFu; }
    __syncthreads();

    // --- Pass 1: stream input via async global->LDS (ASYNCcnt), build bitmap ---
    const float* base = x + (size_t)b * (HW * 4);   // 4 channels per cell, 16 B/cell
    const unsigned int lds0 = (unsigned int)(size_t)&stage[0][tid];
    const unsigned int lds1 = (unsigned int)(size_t)&stage[1][tid];

    {   // prologue: issue step 0
        unsigned int voff = (unsigned int)(tid * 16);
        asm volatile("global_load_async_to_lds_b128 %0, %1, %2"
                     :: "v"(lds0), "v"(voff), "s"(base) : "memory");
    }

    unsigned int foodCnt = 0u;
    unsigned int oppMin  = 0xFFFFFFFFu;

    for (int s = 0; s < STEPS; ++s) {
        if (s + 1 < STEPS) {
            unsigned int voff = (unsigned int)(((s + 1) * TPB + tid) * 16);
            unsigned int ldsn = ((s + 1) & 1) ? lds1 : lds0;
            asm volatile("global_load_async_to_lds_b128 %0, %1, %2"
                         :: "v"(ldsn), "v"(voff), "s"(base) : "memory");
            asm volatile("s_wait_asynccnt 0x1" ::: "memory");   // oldest step landed
        } else {
            asm volatile("s_wait_asynccnt 0x0" ::: "memory");
        }

        // Each lane consumes only its own staged 16B cell (no cross-lane LDS use).
        float4 cell = stage[s & 1][tid];
        asm volatile("s_wait_dscnt 0x0" ::: "memory");  // data in VGPRs before buffer reuse

        bool food = (cell.y == 1.0f);   // channel 1
        bool opp  = (cell.w == 1.0f);   // channel 3

        unsigned int bf = __builtin_amdgcn_ballot_w32(food);
        unsigned int bo = __builtin_amdgcn_ballot_w32(opp);

        if (lane == 0) {
            bitmap[s * (TPB / 32) + wave] = bf;         // covers all 512 words
            foodCnt += (unsigned int)__popc(bf);
            if (bo) {
                unsigned int first = (unsigned int)(s * TPB + (wave << 5))
                                   + (unsigned int)__builtin_ctz(bo);
                oppMin = oppMin < first ? oppMin : first;
            }
        }
    }

    if (lane == 0) {
        atomicMin(&s_opp, oppMin);
        atomicAdd(&s_cnt, foodCnt);
    }
    __syncthreads();

    // --- Pass 2: argmin of exact integer squared distance over food bitmap ---
    const unsigned int oppFirst = (s_opp == 0xFFFFFFFFu) ? 0u : s_opp;
    const int opr = (int)(oppFirst >> 7);
    const int opc = (int)(oppFirst & 127u);

    unsigned int keyMin = 0xFFFFFFFFu;
    for (int w = tid; w < NWORDS; w += TPB) {
        unsigned int m = bitmap[w];
        int basei = w << 5;
        while (m) {
            int bit = __builtin_ctz(m);
            m &= m - 1u;
            int idx = basei + bit;
            int dr = (idx >> 7) - opr;
            int dc = (idx & 127) - opc;
            unsigned int d2 = (unsigned int)(dr * dr + dc * dc);   // <= 32258 < 2^15
            unsigned int key = (d2 << 14) | (unsigned int)idx;     // idx in 14 bits
            keyMin = keyMin < key ? keyMin : key;
        }
    }
    atomicMin(&s_key, keyMin);
    __syncthreads();

    // --- Output: zero both maps for this batch, then place the single 1.0 ---
    float4* g = (float4*)out + (size_t)b * (HW / 4);
    float4* z = (float4*)out + (size_t)(BATCH + b) * (HW / 4);
    const float4 zero = make_float4(0.f, 0.f, 0.f, 0.f);
    for (int i = tid; i < HW / 4; i += TPB) {
        g[i] = zero;
        z[i] = zero;
    }
    __syncthreads();

    if (tid == 0) {
        unsigned int cnt = s_cnt;
        bool oppStart = (opr == 3) && (opc == 6);
        bool useMin = (cnt == 1u) || (cnt > 1u && !oppStart);
        unsigned int tgt = useMin ? (s_key & 0x3FFFu) : 0u;
        if (cnt > 0u) out[(size_t)b * HW + tgt] = 1.0f;
    }
}

extern "C" void kernel_launch(void* const* d_in, const int* in_sizes, int n_in,
                              void* d_out, int out_size, void* d_ws, size_t ws_size,
                              hipStream_t stream) {
    const float* x = (const float*)d_in[0];
    float* out = (float*)d_out;
    hipLaunchKernelGGL(oracle_target_map_kernel, dim3(BATCH), dim3(TPB), 0, stream,
                       x, out);
}